// Attention_57861799412200
// MI455X (gfx1250) — compile-verified
//
#include <hip/hip_runtime.h>

#define B_ 128
#define S_ 4096
#define E_ 256   // ENC
#define D_ 256   // DEC
#define SPLITS 4
#define ROWS_PER_SPLIT (S_ / SPLITS)   // 1024
#define PART_STRIDE 258                // 256 acc + m + l

typedef __attribute__((ext_vector_type(2))) float v2f;
typedef __attribute__((ext_vector_type(8))) float v8f;

// ---------------------------------------------------------------------------
// Kernel 1: proj[128x256] = dh[128x256] @ W^T  (W is ENCxDEC row-major)
// Uses V_WMMA_F32_16X16X4_F32. One wave computes one 16x16 tile of proj.
// A (16x4 f32):  lane L(0..15) holds {K=k, K=k+1}; lane L+16 holds {K=k+2, K=k+3}
// B (4x16 f32):  same K split, lane-low-4-bits = column n
// C/D (16x16):   VGPR j: lanes 0-15 -> row m0+j, lanes 16-31 -> row m0+8+j
// ---------------------------------------------------------------------------
__global__ void __launch_bounds__(256) proj_wmma(const float* __restrict__ dh,
                                                 const float* __restrict__ w,
                                                 float* __restrict__ proj) {
    const int lane = threadIdx.x & 31;
    const int wid  = blockIdx.x * (blockDim.x >> 5) + (threadIdx.x >> 5); // 0..127
    const int m0   = (wid >> 4) * 16;   // 8 row tiles
    const int n0   = (wid & 15) * 16;   // 16 col tiles
    const int half = lane >> 4;         // 0 or 1
    const int l15  = lane & 15;

    const float* arow = dh + (size_t)(m0 + l15) * D_;  // A row (batch m0+l15)
    const float* brow = w  + (size_t)(n0 + l15) * D_;  // W row n -> B column n
    v8f c = {};
#pragma unroll 8
    for (int k = 0; k < D_; k += 4) {
        const int kk = k + 2 * half;
        v2f a, b;
        a[0] = arow[kk]; a[1] = arow[kk + 1];
        b[0] = brow[kk]; b[1] = brow[kk + 1];
        c = __builtin_amdgcn_wmma_f32_16x16x4_f32(false, a, false, b,
                                                  (short)0, c, false, false);
    }
    float* out = proj + (size_t)(m0 + 8 * half) * E_ + n0 + l15;
#pragma unroll
    for (int j = 0; j < 8; ++j)
        out[(size_t)j * E_] = c[j];
}

// ---------------------------------------------------------------------------
// Kernel 2: fused masked-softmax attention, online softmax, one pass over enc.
// Grid: B*SPLITS blocks x 256 threads. Each wave cooperates on one enc row:
// lane holds 8 ENC elements (2x float4, coalesced 1KB/row per wave).
// Masked rows are skipped before loading (saves ~50% HBM traffic).
// Emits per-(batch,split) partial (acc[256], m, l) to workspace.
// ---------------------------------------------------------------------------
__global__ void __launch_bounds__(256) attn_main(const float* __restrict__ enc,
                                                 const float* __restrict__ proj,
                                                 const int*   __restrict__ mask,
                                                 float* __restrict__ part) {
    const int b     = blockIdx.x / SPLITS;
    const int split = blockIdx.x % SPLITS;
    const int lane  = threadIdx.x & 31;
    const int wave  = threadIdx.x >> 5;   // 0..7

    const float* pb = proj + (size_t)b * E_ + lane * 8;
    const float4 p0 = *(const float4*)(pb);
    const float4 p1 = *(const float4*)(pb + 4);

    const float NEG_INF = -__builtin_inff();
    float m = NEG_INF, l = 0.f;
    float4 a0 = make_float4(0.f, 0.f, 0.f, 0.f);
    float4 a1 = make_float4(0.f, 0.f, 0.f, 0.f);

    const int    sBegin  = split * ROWS_PER_SPLIT;
    const size_t encBase = (size_t)b * S_ * E_;

    for (int s0 = sBegin + wave * 32; s0 < sBegin + ROWS_PER_SPLIT; s0 += 32 * 8) {
        const int    mrow = mask[(size_t)b * S_ + s0 + lane];  // 32 masks, 1/lane
        const float* rb   = enc + encBase + (size_t)s0 * E_ + lane * 8;
#pragma unroll 4
        for (int r = 0; r < 32; ++r) {
            if (__shfl(mrow, r, 32) == 0) continue;  // wave-uniform skip, no load
            const float4 e0 = *(const float4*)(rb + (size_t)r * E_);
            const float4 e1 = *(const float4*)(rb + (size_t)r * E_ + 4);
            float partial = e0.x*p0.x + e0.y*p0.y + e0.z*p0.z + e0.w*p0.w
                          + e1.x*p1.x + e1.y*p1.y + e1.z*p1.z + e1.w*p1.w;
#pragma unroll
            for (int off = 16; off > 0; off >>= 1)
                partial += __shfl_xor(partial, off, 32);
            const float score = partial;             // uniform across wave
            if (score <= m) {                        // common case: no rescale
                const float wgt = __expf(score - m);
                l += wgt;
                a0.x += wgt*e0.x; a0.y += wgt*e0.y; a0.z += wgt*e0.z; a0.w += wgt*e0.w;
                a1.x += wgt*e1.x; a1.y += wgt*e1.y; a1.z += wgt*e1.z; a1.w += wgt*e1.w;
            } else {                                 // new max: rescale, wgt == 1
                const float alpha = __expf(m - score);   // 0 when m == -inf
                l = l * alpha + 1.f;
                a0.x = a0.x*alpha + e0.x; a0.y = a0.y*alpha + e0.y;
                a0.z = a0.z*alpha + e0.z; a0.w = a0.w*alpha + e0.w;
                a1.x = a1.x*alpha + e1.x; a1.y = a1.y*alpha + e1.y;
                a1.z = a1.z*alpha + e1.z; a1.w = a1.w*alpha + e1.w;
                m = score;
            }
        }
    }

    // Cross-wave merge through LDS (8 waves -> one partial per block).
    __shared__ float4 sacc[8][64];
    __shared__ float  sm[8], sl[8];
    sacc[wave][lane * 2]     = a0;
    sacc[wave][lane * 2 + 1] = a1;
    if (lane == 0) { sm[wave] = m; sl[wave] = l; }
    __syncthreads();

    const int e = threadIdx.x;      // 256 threads -> one ENC element each
    float M = NEG_INF;
#pragma unroll
    for (int wv = 0; wv < 8; ++wv) M = fmaxf(M, sm[wv]);
    float L = 0.f, A = 0.f;
    const float* sa = (const float*)sacc;
#pragma unroll
    for (int wv = 0; wv < 8; ++wv) {
        const float mw = sm[wv];
        if (mw == NEG_INF) continue;
        const float f = __expf(mw - M);
        L += sl[wv] * f;
        A += sa[wv * 256 + e] * f;
    }
    float* po = part + (size_t)(b * SPLITS + split) * PART_STRIDE;
    po[e] = A;
    if (threadIdx.x == 0) { po[256] = M; po[257] = L; }
}

// ---------------------------------------------------------------------------
// Kernel 3: merge SPLITS partials per batch; divide by global softmax sum.
// ---------------------------------------------------------------------------
__global__ void __launch_bounds__(256) attn_combine(const float* __restrict__ part,
                                                    float* __restrict__ out) {
    const int b = blockIdx.x;
    const int e = threadIdx.x;
    const float* pb = part + (size_t)b * SPLITS * PART_STRIDE;
    const float NEG_INF = -__builtin_inff();
    float M = NEG_INF;
#pragma unroll
    for (int sp = 0; sp < SPLITS; ++sp)
        M = fmaxf(M, pb[sp * PART_STRIDE + 256]);
    float L = 0.f, A = 0.f;
#pragma unroll
    for (int sp = 0; sp < SPLITS; ++sp) {
        const float msp = pb[sp * PART_STRIDE + 256];
        if (msp == NEG_INF) continue;
        const float f = __expf(msp - M);
        L += pb[sp * PART_STRIDE + 257] * f;
        A += pb[sp * PART_STRIDE + e] * f;
    }
    out[(size_t)b * E_ + e] = A / L;   // >=1 valid position guaranteed -> L > 0
}

// ---------------------------------------------------------------------------
extern "C" void kernel_launch(void* const* d_in, const int* in_sizes, int n_in,
                              void* d_out, int out_size, void* d_ws, size_t ws_size,
                              hipStream_t stream) {
    const float* dh   = (const float*)d_in[0];   // (B, DEC)
    const float* enc  = (const float*)d_in[1];   // (B, S, ENC)
    const float* w    = (const float*)d_in[2];   // (ENC, DEC)
    const int*   mask = (const int*)d_in[3];     // (B, S)
    float* out  = (float*)d_out;                 // (B, ENC)

    float* proj = (float*)d_ws;                  // 128*256 floats
    float* part = proj + B_ * E_;                // B*SPLITS*258 floats

    proj_wmma   <<<dim3(16),          dim3(256), 0, stream>>>(dh, w, proj);
    attn_main   <<<dim3(B_ * SPLITS), dim3(256), 0, stream>>>(enc, proj, mask, part);
    attn_combine<<<dim3(B_),          dim3(256), 0, stream>>>(part, out);
}